// toyModel_77893526880800
// MI455X (gfx1250) — compile-verified
//
#include <hip/hip_runtime.h>

typedef __attribute__((ext_vector_type(2))) float v2f;
typedef __attribute__((ext_vector_type(8))) float v8f;

#define HDEG 8
#define DIM 64
#define LDEG 256
#define SEQ 8192
#define BATCH 32
#define NSPLIT 8
#define CHUNK (SEQ / NSPLIT)        // 1024 positions per block
#define ROUND_T 128                 // 8 waves * 16 t-values per round
#define NROUND (CHUNK / ROUND_T)    // 8
#define XT_STRIDE 68                // padded row stride (floats) -> conflict-free ds_load_b64
#define DB_STRIDE 144               // 16 halo + 128 columns

// ---------------------------------------------------------------------------
// Kernel 1: per (batch, chunk) -> chunk-local softmax partials in workspace
// ---------------------------------------------------------------------------
__global__ __launch_bounds__(256) void poly_attn_phase1(
    const float* __restrict__ x, const float* __restrict__ coeff,
    float* __restrict__ ws)
{
  const int tid  = threadIdx.x;
  const int lane = tid & 31;
  const int wv   = tid >> 5;            // wave id 0..7
  const int b    = blockIdx.x;
  const int sp   = blockIdx.y;
  const int T0   = sp * CHUNK;

  extern __shared__ float smem[];
  float* c2s    = smem;                         // 256
  float* logits = c2s + LDEG;                   // 1024
  float* dbuf   = logits + CHUNK;               // 8*144
  float* xtile  = dbuf + HDEG * DB_STRIDE;      // 8 waves * 16*68
  float* red    = xtile + 8 * 16 * XT_STRIDE;   // 256 scratch

  // coeff^2 into LDS; zero the d ring (halo must be 0 for T0==0 / t<0 cases)
  {
    float cv = coeff[tid];
    c2s[tid] = cv * cv;
  }
  for (int i = tid; i < HDEG * DB_STRIDE; i += 256) dbuf[i] = 0.0f;
  __syncthreads();

  float sumc2 = 0.f;
  for (int l = 0; l < LDEG; ++l) sumc2 += c2s[l];
  const float scaleA = 0.01f / sumc2;           // A / sum(c2)

  const float* xb = x + (size_t)b * SEQ * DIM;

  // A fragments (Q = last 8 rows of x[b], zero-padded to 16) held in regs.
  // 32-bit A 16x4 layout: lane = (k>=2)*16 + m ; vgpr0=K even, vgpr1=K odd pair.
  const int m  = lane & 15;
  const int kb = (lane >> 4) * 2;
  v2f afrag[16];
  if (m < HDEG) {
    const float* qrow = xb + (size_t)(SEQ - 1 - m) * DIM;
#pragma unroll
    for (int s = 0; s < 16; ++s) {
      afrag[s].x = qrow[4 * s + kb];
      afrag[s].y = qrow[4 * s + kb + 1];
    }
  } else {
#pragma unroll
    for (int s = 0; s < 16; ++s) { afrag[s].x = 0.f; afrag[s].y = 0.f; }
  }

  const int n = lane & 15;

  // Wave-scope: load 16 rows of x into this wave's LDS tile, run 16 fp32 WMMAs
  // (K=4 each -> K=64). D row h (h<8) = d_h over 16 consecutive t.
  auto compute_tile = [&](int t0, float* xt) -> v8f {
#pragma unroll
    for (int i = 0; i < 8; ++i) {
      int idx = lane + 32 * i;        // 256 float4 slots = 16 rows x 16 quads
      int row = idx >> 4;
      int c4  = idx & 15;
      const float* gsrc = xb + (size_t)(t0 + row) * DIM + c4 * 4;
      // prefetch next round's tile into cache hierarchy
      if (t0 + ROUND_T + row < SEQ)
        __builtin_prefetch(gsrc + (size_t)ROUND_T * DIM, 0, 1);
      const float4 val = *(const float4*)gsrc;
      float* dst = xt + row * XT_STRIDE + c4 * 4;
      dst[0] = val.x; dst[1] = val.y; dst[2] = val.z; dst[3] = val.w;
    }
    // B 4x16 fragment: lane = (k>=2)*16 + n ; b = {X[t0+n][4s+kb], X[t0+n][4s+kb+1]}
    const float* brow = xt + n * XT_STRIDE + kb;
    v8f c = {0.f, 0.f, 0.f, 0.f, 0.f, 0.f, 0.f, 0.f};
#pragma unroll
    for (int s = 0; s < 16; ++s) {
      v2f bf = *(const v2f*)(brow + 4 * s);
      c = __builtin_amdgcn_wmma_f32_16x16x4_f32(
          /*neg_a=*/false, afrag[s], /*neg_b=*/false, bf,
          /*c_mod=*/(short)0, c, /*reuse_a=*/false, /*reuse_b=*/false);
    }
    return c;
  };

  // Lookback tile for the halo (d values for t in [T0-16, T0)).
  if (T0 > 0 && wv == 0) {
    v8f d = compute_tile(T0 - 16, xtile);
    if (lane < 16) {
#pragma unroll
      for (int h = 0; h < HDEG; ++h) dbuf[h * DB_STRIDE + lane] = d[h];
    }
  }
  __syncthreads();

  for (int r = 0; r < NROUND; ++r) {
    const int t0 = T0 + r * ROUND_T + wv * 16;
    float* xt = xtile + wv * 16 * XT_STRIDE;
    v8f d = compute_tile(t0, xt);
    if (lane < 16) {
#pragma unroll
      for (int h = 0; h < HDEG; ++h)
        dbuf[h * DB_STRIDE + 16 + wv * 16 + n] = d[h];
    }
    __syncthreads();

    // Logits: S = sum_l c2[l] * prod_{h in S_l} (c_h + 1e-24), factored 16x16.
    if (lane < 16) {
      const int basecol = wv * 16 + n + 15;   // = 16 + (t-1-h) - row_base + h
      float a[HDEG];
#pragma unroll
      for (int h = 0; h < HDEG; ++h)
        a[h] = dbuf[h * DB_STRIDE + (basecol - h)] + 1e-24f;

      float Plo[16], Phi[16];
      Plo[0] = 1.f; Phi[0] = 1.f;
#pragma unroll
      for (int k = 0; k < 4; ++k) {
        const int half = 1 << k;
#pragma unroll
        for (int j = 0; j < 8; ++j) {
          if (j < half) {
            Plo[half + j] = Plo[j] * a[7 - k];   // low 4 bits of l  -> h = 7-k
            Phi[half + j] = Phi[j] * a[3 - k];   // high 4 bits of l -> h = 3-k
          }
        }
      }
      float S = 0.f;
#pragma unroll
      for (int u = 0; u < 16; ++u) {
        float inner = 0.f;
#pragma unroll
        for (int w2 = 0; w2 < 16; ++w2)
          inner = fmaf(c2s[16 * u + w2], Plo[w2], inner);
        S = fmaf(Phi[u], inner, S);
      }
      logits[r * ROUND_T + wv * 16 + n] = S * scaleA;
    }
    __syncthreads();

    // Slide halo: last 16 columns -> [0,16)
    if (tid < HDEG * 16) {
      int h = tid >> 4, nn = tid & 15;
      dbuf[h * DB_STRIDE + nn] = dbuf[h * DB_STRIDE + 128 + nn];
    }
    __syncthreads();
  }

  // ---- chunk-local softmax partials ----
  float lm = -3.4e38f;
  for (int i = tid; i < CHUNK; i += 256) lm = fmaxf(lm, logits[i]);
  red[tid] = lm; __syncthreads();
  for (int sr = 128; sr > 0; sr >>= 1) {
    if (tid < sr) red[tid] = fmaxf(red[tid], red[tid + sr]);
    __syncthreads();
  }
  const float mloc = red[0];
  __syncthreads();

  float ls = 0.f;
  for (int i = tid; i < CHUNK; i += 256) {
    float e = __expf(logits[i] - mloc);
    logits[i] = e;
    ls += e;
  }
  red[tid] = ls; __syncthreads();
  for (int sr = 128; sr > 0; sr >>= 1) {
    if (tid < sr) red[tid] += red[tid + sr];
    __syncthreads();
  }
  const float sloc = red[0];
  __syncthreads();

  // acc[dim] = sum_t e_t * x[b, T0+t, dim]  (chunk re-read is L2-resident)
  const int dimid = tid & 63;
  const int slice = tid >> 6;       // 4 slices of CHUNK/4
  float acc = 0.f;
  for (int i = 0; i < CHUNK / 4; ++i) {
    int t = slice * (CHUNK / 4) + i;
    acc = fmaf(logits[t], xb[(size_t)(T0 + t) * DIM + dimid], acc);
  }
  red[tid] = acc; __syncthreads();
  if (tid < 64) {
    float a4 = red[tid] + red[tid + 64] + red[tid + 128] + red[tid + 192];
    float* wp = ws + ((size_t)b * NSPLIT + sp) * 66;
    if (tid == 0) { wp[0] = mloc; wp[1] = sloc; }
    wp[2 + tid] = a4;
  }
}

// ---------------------------------------------------------------------------
// Kernel 2: merge NSPLIT partials per batch (log-sum-exp combine)
// ---------------------------------------------------------------------------
__global__ __launch_bounds__(64) void poly_attn_phase2(
    const float* __restrict__ ws, float* __restrict__ out)
{
  const int b = blockIdx.x;
  const int dimid = threadIdx.x;
  const float* wp = ws + (size_t)b * NSPLIT * 66;
  float mg = -3.4e38f;
  for (int i = 0; i < NSPLIT; ++i) mg = fmaxf(mg, wp[i * 66 + 0]);
  float denom = 0.f, num = 0.f;
  for (int i = 0; i < NSPLIT; ++i) {
    float sc = __expf(wp[i * 66 + 0] - mg);
    denom += wp[i * 66 + 1] * sc;
    num = fmaf(wp[i * 66 + 2 + dimid], sc, num);
  }
  out[b * DIM + dimid] = num / denom;
}

extern "C" void kernel_launch(void* const* d_in, const int* in_sizes, int n_in,
                              void* d_out, int out_size, void* d_ws, size_t ws_size,
                              hipStream_t stream) {
  const float* x     = (const float*)d_in[0];   // (32, 8192, 64) f32
  const float* coeff = (const float*)d_in[1];   // (1, 256) f32
  float* out = (float*)d_out;                   // (32, 1, 64) f32
  float* wsf = (float*)d_ws;                    // 32*8*66 floats of scratch

  const size_t smem =
      (size_t)(LDEG + CHUNK + HDEG * DB_STRIDE + 8 * 16 * XT_STRIDE + 256) *
      sizeof(float);                            // ~44.5 KB

  dim3 grid1(BATCH, NSPLIT);
  poly_attn_phase1<<<grid1, dim3(256), smem, stream>>>(x, coeff, wsf);
  poly_attn_phase2<<<dim3(BATCH), dim3(64), 0, stream>>>(wsf, out);
}